// GCNBlock_7456063226247
// MI455X (gfx1250) — compile-verified
//
#include <hip/hip_runtime.h>

#define N_NODES 100000
#define N_EDGES 3200000
#define D 128

typedef __attribute__((ext_vector_type(2))) float v2f;
typedef __attribute__((ext_vector_type(8))) float v8f;

// ---- degree: deg[n] = 1 (self loop) + sum_{e: dst=n} ew[e] ----
__global__ void k_init_deg(float* __restrict__ deg) {
    int i = blockIdx.x * blockDim.x + threadIdx.x;
    if (i < N_NODES) deg[i] = 1.0f;
}

__global__ void k_accum_deg(const float* __restrict__ ew, const int* __restrict__ dst,
                            float* __restrict__ deg) {
    int e = blockIdx.x * blockDim.x + threadIdx.x;
    if (e < N_EDGES) atomicAdd(&deg[dst[e]], ew[e]);
}

__global__ void k_dinv(const float* __restrict__ deg, float* __restrict__ dinv) {
    int i = blockIdx.x * blockDim.x + threadIdx.x;
    if (i < N_NODES) {
        float d = deg[i];
        dinv[i] = (d > 0.0f) ? rsqrtf(d) : 0.0f;
    }
}

// ---- h = x @ W^T via V_WMMA_F32_16X16X4_F32 (exact fp32) ----
// Block = 256 threads = 8 waves. Block b owns M-tile rows [16b,16b+16);
// wave w owns N-tile cols [16w,16w+16). K=128 accumulated in 32 steps of 4.
// A(16x4 f32) layout: lane<16 -> row=lane, K={0,1}; lane>=16 -> row=lane-16, K={2,3}
// B mirrored; C/D: VGPR r: lanes0-15 -> (row r, col lane), lanes16-31 -> (row r+8, col lane-16)
__global__ void __launch_bounds__(256) k_gemm(const float* __restrict__ x,
                                              const float* __restrict__ w,
                                              float* __restrict__ h) {
    const int wave = threadIdx.x >> 5;
    const int lane = threadIdx.x & 31;
    const int m0 = blockIdx.x << 4;
    const int n0 = wave << 4;
    const int l15 = lane & 15;
    const int hi  = lane >> 4;            // 0 or 1 -> K offset 0 or 2

    const float* arow = x + (size_t)(m0 + l15) * D + 2 * hi;   // x[m][k..k+1]
    const float* brow = w + (size_t)(n0 + l15) * D + 2 * hi;   // W[n][k..k+1] (B = W^T in (k,n))

    v8f c = {};
#pragma unroll
    for (int k = 0; k < D; k += 4) {
        v2f a = *(const v2f*)(arow + k);
        v2f b = *(const v2f*)(brow + k);
        c = __builtin_amdgcn_wmma_f32_16x16x4_f32(
                /*neg_a=*/false, a, /*neg_b=*/false, b,
                /*c_mod=*/(short)0, c, /*reuse_a=*/false, /*reuse_b=*/false);
    }

    const int col = n0 + l15;
    const int rowbase = m0 + hi * 8;
#pragma unroll
    for (int r = 0; r < 8; ++r)
        h[(size_t)(rowbase + r) * D + col] = c[r];
}

// ---- self-loop contribution: acc = h * dinv^2 (initializes d_out accumulator) ----
__global__ void k_selfloop(const float* __restrict__ h, const float* __restrict__ dinv,
                           float* __restrict__ acc) {
    int i = blockIdx.x * blockDim.x + threadIdx.x;
    if (i < N_NODES * D) {
        int n = i >> 7;
        float di = dinv[n];
        acc[i] = h[i] * (di * di);
    }
}

// ---- edge scatter: one wave32 per edge, float4 gather + 4 f32 atomics per lane ----
__global__ void __launch_bounds__(256) k_edges(const float* __restrict__ h,
                                               const float* __restrict__ dinv,
                                               const float* __restrict__ ew,
                                               const int* __restrict__ src,
                                               const int* __restrict__ dst,
                                               float* __restrict__ acc) {
    const int lane = threadIdx.x & 31;
    const int e = blockIdx.x * (blockDim.x >> 5) + (threadIdx.x >> 5);
    if (e >= N_EDGES) return;

    const int s = src[e];
    const int d = dst[e];
    const float norm = dinv[s] * ew[e] * dinv[d];

    const float4 v = ((const float4*)(h + (size_t)s * D))[lane];
    float* a = acc + (size_t)d * D + lane * 4;
    atomicAdd(a + 0, v.x * norm);
    atomicAdd(a + 1, v.y * norm);
    atomicAdd(a + 2, v.z * norm);
    atomicAdd(a + 3, v.w * norm);
}

// ---- out = relu(acc + bias) + x ----
__global__ void k_final(float* __restrict__ out, const float* __restrict__ x,
                        const float* __restrict__ bias) {
    int i = blockIdx.x * blockDim.x + threadIdx.x;
    if (i < N_NODES * D) {
        float v = out[i] + bias[i & (D - 1)];
        v = fmaxf(v, 0.0f);
        out[i] = v + x[i];
    }
}

extern "C" void kernel_launch(void* const* d_in, const int* in_sizes, int n_in,
                              void* d_out, int out_size, void* d_ws, size_t ws_size,
                              hipStream_t stream) {
    const float* x    = (const float*)d_in[0];   // [100000,128]
    const float* w    = (const float*)d_in[1];   // [128,128]
    const float* bias = (const float*)d_in[2];   // [128]
    const float* ew   = (const float*)d_in[3];   // [3200000]
    const int*   eidx = (const int*)d_in[4];     // [2,3200000]
    const int* src = eidx;
    const int* dst = eidx + N_EDGES;
    float* out = (float*)d_out;

    // workspace: deg[100096] | dinv[100096] | h[100000*128]  (~52 MB)
    float* deg  = (float*)d_ws;
    float* dinv = deg + 100096;
    float* h    = dinv + 100096;

    k_init_deg <<<(N_NODES + 255) / 256, 256, 0, stream>>>(deg);
    k_accum_deg<<<(N_EDGES + 255) / 256, 256, 0, stream>>>(ew, dst, deg);
    k_dinv     <<<(N_NODES + 255) / 256, 256, 0, stream>>>(deg, dinv);
    k_gemm     <<<N_NODES / 16, 256, 0, stream>>>(x, w, h);
    k_selfloop <<<(N_NODES * D + 255) / 256, 256, 0, stream>>>(h, dinv, out);
    k_edges    <<<(N_EDGES + 7) / 8, 256, 0, stream>>>(h, dinv, ew, src, dst, out);
    k_final    <<<(N_NODES * D + 255) / 256, 256, 0, stream>>>(out, x, bias);
}